// BoneLengthLoss_91139206021210
// MI455X (gfx1250) — compile-verified
//
#include <hip/hip_runtime.h>
#include <stdint.h>

#define NKPT 37
#define NBONES 32
#define FPP (NKPT * 3)            // 111 floats per pose
#define TILE_B 8                  // batch elements per tile (one per wave)
#define TILE_F (TILE_B * FPP)     // 888 floats
#define TILE_BYTES (TILE_F * 4)   // 3552 bytes = 222 x 16B chunks
#define CHUNKS (TILE_BYTES / 16)  // 222 (< 256 -> every copier thread: exactly 2 async ops/tile)
#define THREADS 256
#define NBUF 4                    // 4-deep async pipeline (3 tiles of prefetch in flight)
#define NBLOCKS 4096

__constant__ unsigned char c_j1[NBONES] = {
    1, 1, 1, 2, 3, 11, 11, 12, 13, 14, 15, 16, 12, 18, 20, 13,
    19, 21, 16, 16, 24, 25, 24, 27, 29, 25, 28, 30, 17, 33, 34, 35};
__constant__ unsigned char c_j2[NBONES] = {
    2, 3, 4, 5, 6, 12, 13, 14, 14, 15, 16, 17, 18, 20, 22, 19,
    21, 23, 24, 25, 26, 26, 27, 29, 31, 28, 30, 32, 33, 34, 35, 36};

// gfx1250 async global->LDS copy, 16B per active lane. Tracked on ASYNCcnt.
__device__ __forceinline__ void async_copy_b128(uint32_t lds_off, uint64_t gaddr) {
    asm volatile("global_load_async_to_lds_b128 %0, %1, off"
                 :
                 : "v"(lds_off), "v"(gaddr)
                 : "memory");
}

// Wait until at most `n` async ops remain in flight (in-order completion).
// n is always 2 * (#prefetched tiles still allowed in flight) in {0,2,4,6}.
__device__ __forceinline__ void wait_async(int n) {
    switch (n) {
        case 0: asm volatile("s_wait_asynccnt 0" ::: "memory"); break;
        case 2: asm volatile("s_wait_asynccnt 2" ::: "memory"); break;
        case 4: asm volatile("s_wait_asynccnt 4" ::: "memory"); break;
        default: asm volatile("s_wait_asynccnt 6" ::: "memory"); break;
    }
}

__global__ __launch_bounds__(THREADS) void bone_loss_partial(
    const float* __restrict__ pred, const float* __restrict__ ref,
    const unsigned char* __restrict__ mask, float* __restrict__ partial,
    int tiles_per_block) {
    __shared__ __align__(16) float s_pred[NBUF][TILE_F];
    __shared__ __align__(16) float s_ref[NBUF][TILE_F];
    __shared__ float s_wn[THREADS / 32];
    __shared__ float s_wd[THREADS / 32];

    const int tid  = threadIdx.x;
    const int wave = tid >> 5;
    const int lane = tid & 31;

    // one bone per lane (32 bones == wave32)
    const int kj1 = c_j1[lane];
    const int kj2 = c_j2[lane];
    const int a1  = kj1 * 3;
    const int a2  = kj2 * 3;

    const int T  = tiles_per_block;
    const int t0 = blockIdx.x * T;

    // LDS byte offsets of the buffers (low 32 bits of flat addr == LDS offset)
    const uint32_t lp = (uint32_t)(uintptr_t)(&s_pred[0][0]);
    const uint32_t lr = (uint32_t)(uintptr_t)(&s_ref[0][0]);

    const uint32_t coff   = (uint32_t)tid * 16u;
    const bool     copier = (tid < CHUNKS);
    const uint64_t gpred  = (uint64_t)(uintptr_t)pred;
    const uint64_t gref   = (uint64_t)(uintptr_t)ref;

    // prologue: stage up to NBUF-1 tiles ahead
    for (int k = 0; k < NBUF - 1 && k < T; ++k) {
        if (copier) {
            const uint32_t nb   = (uint32_t)k * TILE_BYTES;
            const uint64_t goff = (uint64_t)(t0 + k) * TILE_BYTES + coff;
            async_copy_b128(lp + nb + coff, gpred + goff);
            async_copy_b128(lr + nb + coff, gref + goff);
        }
    }

    float num = 0.0f, den = 0.0f;

    for (int i = 0; i < T; ++i) {
        const int buf = i & (NBUF - 1);

        // prefetch tile i+NBUF-1 into the buffer last read at iteration i-1
        // (trailing barrier of iter i-1 guarantees it is free)
        if (i + NBUF - 1 < T) {
            if (copier) {
                const uint32_t nb   = (uint32_t)((i + NBUF - 1) & (NBUF - 1)) * TILE_BYTES;
                const uint64_t goff = (uint64_t)(t0 + i + NBUF - 1) * TILE_BYTES + coff;
                async_copy_b128(lp + nb + coff, gpred + goff);
                async_copy_b128(lr + nb + coff, gref + goff);
            }
        }
        // tiles still allowed in flight after this wait: i+1 .. min(T-1, i+NBUF-1)
        int inflight = T - 1 - i;
        if (inflight > NBUF - 1) inflight = NBUF - 1;
        wait_async(2 * inflight);  // in-order => tile i's ops are complete
        __syncthreads();           // all waves' async writes for tile i visible

        // compute: wave handles batch element b, lane handles its bone
        const float* pp = &s_pred[buf][wave * FPP];
        const float* rr = &s_ref[buf][wave * FPP];
        const long long b = (long long)(t0 + i) * TILE_B + wave;

        const unsigned char* mb = mask + b * NKPT;
        const unsigned char  m1 = mb[kj1];
        const unsigned char  m2 = mb[kj2];

        const float pdx = pp[a2 + 0] - pp[a1 + 0];
        const float pdy = pp[a2 + 1] - pp[a1 + 1];
        const float pdz = pp[a2 + 2] - pp[a1 + 2];
        const float rdx = rr[a2 + 0] - rr[a1 + 0];
        const float rdy = rr[a2 + 1] - rr[a1 + 1];
        const float rdz = rr[a2 + 2] - rr[a1 + 2];
        const float pl  = sqrtf(pdx * pdx + pdy * pdy + pdz * pdz);
        const float rl  = sqrtf(rdx * rdx + rdy * rdy + rdz * rdz);

        const float v = (float)((m1 & m2) & 1);
        const float d = pl - rl;
        num += v * d * d;
        den += v;

        __syncthreads();  // everyone done reading buf before it is refilled
    }

    // wave32 butterfly reduction
    for (int off = 16; off > 0; off >>= 1) {
        num += __shfl_xor(num, off, 32);
        den += __shfl_xor(den, off, 32);
    }
    if (lane == 0) { s_wn[wave] = num; s_wd[wave] = den; }
    __syncthreads();
    if (tid == 0) {
        float n = 0.0f, dd = 0.0f;
        for (int w = 0; w < THREADS / 32; ++w) { n += s_wn[w]; dd += s_wd[w]; }
        partial[2 * blockIdx.x + 0] = n;
        partial[2 * blockIdx.x + 1] = dd;
    }
}

__global__ __launch_bounds__(256) void bone_loss_finalize(
    const float* __restrict__ partial, float* __restrict__ out, int nblocks) {
    __shared__ double sn[256];
    __shared__ double sd[256];
    double n = 0.0, d = 0.0;
    for (int i = threadIdx.x; i < nblocks; i += 256) {
        n += (double)partial[2 * i + 0];
        d += (double)partial[2 * i + 1];
    }
    sn[threadIdx.x] = n;
    sd[threadIdx.x] = d;
    __syncthreads();
    for (int s = 128; s > 0; s >>= 1) {
        if (threadIdx.x < s) {
            sn[threadIdx.x] += sn[threadIdx.x + s];
            sd[threadIdx.x] += sd[threadIdx.x + s];
        }
        __syncthreads();
    }
    if (threadIdx.x == 0) out[0] = (float)(sn[0] / sd[0]);
}

extern "C" void kernel_launch(void* const* d_in, const int* in_sizes, int n_in,
                              void* d_out, int out_size, void* d_ws, size_t ws_size,
                              hipStream_t stream) {
    const float*         pred = (const float*)d_in[0];
    const float*         ref  = (const float*)d_in[1];
    const unsigned char* mask = (const unsigned char*)d_in[2];  // jnp.bool_ -> 1 byte
    float* out     = (float*)d_out;
    float* partial = (float*)d_ws;

    const int batch = in_sizes[0] / FPP;  // 524288
    const int tiles = batch / TILE_B;     // 65536
    int blocks = NBLOCKS;
    if (blocks > tiles) blocks = tiles;
    const int tpb = tiles / blocks;       // 16

    bone_loss_partial<<<blocks, THREADS, 0, stream>>>(pred, ref, mask, partial, tpb);
    bone_loss_finalize<<<1, 256, 0, stream>>>(partial, out, blocks);
}